// HAN_34651796144563
// MI455X (gfx1250) — compile-verified
//
#include <hip/hip_runtime.h>
#include <hip/hip_bf16.h>

#define NNODES 100000
#define NEDGES 400000
#define HIDF   128
#define NHEAD  8
#define HEADD  16

typedef __attribute__((ext_vector_type(2))) float v2f;
typedef __attribute__((ext_vector_type(4))) float v4f;
typedef __attribute__((ext_vector_type(8))) float v8f;
typedef __attribute__((ext_vector_type(4))) int   v4i;

#ifndef __has_builtin
#define __has_builtin(x) 0
#endif
#if __has_builtin(__builtin_amdgcn_global_load_async_to_lds_b128) && \
    __has_builtin(__builtin_amdgcn_s_wait_asynccnt)
#define HAVE_ASYNC_LDS 1
#else
#define HAVE_ASYNC_LDS 0
#endif

#define AS1 __attribute__((address_space(1)))
#define AS3 __attribute__((address_space(3)))

// ---------- monotone float<->uint mapping for atomic max on floats ----------
__device__ __forceinline__ unsigned f2mono(float f) {
  unsigned b = __float_as_uint(f);
  return (b & 0x80000000u) ? ~b : (b | 0x80000000u);
}
__device__ __forceinline__ float mono2f(unsigned u) {
  unsigned b = (u & 0x80000000u) ? (u & 0x7FFFFFFFu) : ~u;
  return __uint_as_float(b);
}

// ---------------------------------------------------------------------------
// Stage one 128(K) x 16(col) fp32 weight slice (8 KB) into LDS, shared by all
// 8 waves of the block. Uses CDNA5 async global->LDS (ASYNCcnt) when the
// builtins are declared; falls back to b128 load + ds_store otherwise.
// 512 x 16B chunks, 2 per thread. Must be called by ALL threads (barrier).
// ---------------------------------------------------------------------------
__device__ __forceinline__ void stage_b_tile(const float* __restrict__ W, int col0,
                                             float* lds_b /*2048 floats*/) {
  const int t = threadIdx.x;
#pragma unroll
  for (int i = 0; i < 2; ++i) {
    int chunk = t * 2 + i;             // 0..511
    int k = chunk >> 2;                // K row 0..127
    int c4 = (chunk & 3) << 2;         // col sub-offset 0,4,8,12
    const float* g = W + (size_t)k * HIDF + col0 + c4;
    float* l = lds_b + chunk * 4;
#if HAVE_ASYNC_LDS
    __builtin_amdgcn_global_load_async_to_lds_b128(
        (AS1 v4i*)(const v4i*)g, (AS3 v4i*)(v4i*)l, 0, 0);
#else
    *(v4f*)l = *(const v4f*)g;
#endif
  }
#if HAVE_ASYNC_LDS
  __builtin_amdgcn_s_wait_asynccnt(0);
#endif
  __syncthreads();
}

// ---------------------------------------------------------------------------
// Projection GEMM: H[M,128] = X[M,128] @ W[128,128] + bias, fp32 WMMA 16x16x4.
// Block = 256 threads (8 waves) covering 128 rows x 16 cols:
//   grid.x -> 8-row-tile groups, grid.y -> column tile (0..7).
// Each wave owns a distinct 16-row tile (A loads have zero redundancy);
// the B slice is LDS-staged once per block.
// ---------------------------------------------------------------------------
__global__ void proj_gemm(const float* __restrict__ X, const float* __restrict__ W,
                          const float* __restrict__ bias, float* __restrict__ Hout) {
  __shared__ float lds_b[HIDF * 16];
  const int col0 = blockIdx.y * 16;
  stage_b_tile(W, col0, lds_b);

  const int lane = threadIdx.x & 31;
  const int wave = threadIdx.x >> 5;
  const int row0 = (blockIdx.x * 8 + wave) * 16;
  if (row0 >= NNODES) return;          // wave-uniform; after the barrier

  const int mloc = lane & 15;          // row within tile (A), col within tile (B/C)
  const int koff = (lane >> 4) * 2;    // K sub-pair held by hi/lo half-wave
  v8f acc = {};
  const float* Arow = X + (size_t)(row0 + mloc) * HIDF;
#pragma unroll
  for (int k = 0; k < HIDF; k += 4) {
    v2f a, b;
    a.x = Arow[k + koff];
    a.y = Arow[k + koff + 1];
    b.x = lds_b[(k + koff) * 16 + mloc];
    b.y = lds_b[(k + koff + 1) * 16 + mloc];
    acc = __builtin_amdgcn_wmma_f32_16x16x4_f32(false, a, false, b, (short)0, acc,
                                                false, false);
  }
  const int colw = col0 + mloc;
  const float bv = bias[colw];
  const int rbase = row0 + (lane >> 4) * 8;   // C layout: vgpr j -> M=j (+8 hi lanes)
#pragma unroll
  for (int j = 0; j < 8; ++j)
    Hout[(size_t)(rbase + j) * HIDF + colw] = acc[j] + bv;
}

// ---------------------------------------------------------------------------
// Semantic GEMM+reduce: colsum[f] += sum_rows tanh( relu(O) @ Wk + bk )[:,f]
// Same tiling; tanh output never materialized.
// ---------------------------------------------------------------------------
__global__ void semantic_colsum(const float* __restrict__ O, const float* __restrict__ Wk,
                                const float* __restrict__ bk, float* __restrict__ colsum) {
  __shared__ float lds_b[HIDF * 16];
  const int col0 = blockIdx.y * 16;
  stage_b_tile(Wk, col0, lds_b);

  const int lane = threadIdx.x & 31;
  const int wave = threadIdx.x >> 5;
  const int row0 = (blockIdx.x * 8 + wave) * 16;
  if (row0 >= NNODES) return;

  const int mloc = lane & 15;
  const int koff = (lane >> 4) * 2;
  v8f acc = {};
  const float* Arow = O + (size_t)(row0 + mloc) * HIDF;
#pragma unroll
  for (int k = 0; k < HIDF; k += 4) {
    v2f a, b;
    a.x = fmaxf(Arow[k + koff], 0.0f);          // ReLU folded into A fragment load
    a.y = fmaxf(Arow[k + koff + 1], 0.0f);
    b.x = lds_b[(k + koff) * 16 + mloc];
    b.y = lds_b[(k + koff + 1) * 16 + mloc];
    acc = __builtin_amdgcn_wmma_f32_16x16x4_f32(false, a, false, b, (short)0, acc,
                                                false, false);
  }
  const int colw = col0 + mloc;
  const float bv = bk[colw];
  float partial = 0.0f;
#pragma unroll
  for (int j = 0; j < 8; ++j) partial += tanhf(acc[j] + bv);
  atomicAdd(&colsum[colw], partial);
}

// ---------------------------------------------------------------------------
// Per-(node,head) attention logit halves: out[n,h] = sum_d H[n,h,d]*att[h,d]
// ---------------------------------------------------------------------------
__global__ void node_attn(const float* __restrict__ Hm, const float* __restrict__ att,
                          float* __restrict__ outp) {
  int idx = blockIdx.x * blockDim.x + threadIdx.x;
  if (idx >= NNODES * NHEAD) return;
  int node = idx >> 3, hh = idx & 7;
  const float* hp = Hm + (size_t)node * HIDF + hh * HEADD;
  const float* ap = att + hh * HEADD;
  float s = 0.0f;
#pragma unroll
  for (int d = 0; d < HEADD; ++d) s += hp[d] * ap[d];
  outp[idx] = s;
}

// ---------------------------------------------------------------------------
// Edge pass 1: logit + leaky-relu, store per-edge, segment-max via atomic.
// ---------------------------------------------------------------------------
__global__ void edge_pass1(const int* __restrict__ src, const int* __restrict__ dst,
                           const float* __restrict__ as_, const float* __restrict__ ad_,
                           float* __restrict__ ea, unsigned* __restrict__ amax) {
  int e = blockIdx.x * blockDim.x + threadIdx.x;
  if (e >= NEDGES) return;
  int s = src[e], d = dst[e];
#pragma unroll
  for (int h = 0; h < NHEAD; ++h) {
    float v = as_[s * NHEAD + h] + ad_[d * NHEAD + h];
    v = (v > 0.0f) ? v : 0.2f * v;              // leaky relu, slope 0.2
    ea[(size_t)e * NHEAD + h] = v;
    atomicMax(&amax[d * NHEAD + h], f2mono(v));
  }
}

// Edge pass 2: ex = exp(a - segmax), store, segment-sum denom.
__global__ void edge_pass2(const int* __restrict__ dst, const unsigned* __restrict__ amax,
                           float* __restrict__ ea, float* __restrict__ denom) {
  int e = blockIdx.x * blockDim.x + threadIdx.x;
  if (e >= NEDGES) return;
  int d = dst[e];
#pragma unroll
  for (int h = 0; h < NHEAD; ++h) {
    float m = mono2f(amax[d * NHEAD + h]);
    float ex = __expf(ea[(size_t)e * NHEAD + h] - m);
    ea[(size_t)e * NHEAD + h] = ex;
    atomicAdd(&denom[d * NHEAD + h], ex);
  }
}

// Edge pass 3: weighted scatter-add of source features. 1 thread per (edge,f).
__global__ void edge_pass3(const int* __restrict__ src, const int* __restrict__ dst,
                           const float* __restrict__ ea, const float* __restrict__ denom,
                           const float* __restrict__ Hsrc, float* __restrict__ O) {
  int idx = blockIdx.x * blockDim.x + threadIdx.x;
  if (idx >= NEDGES * HIDF) return;
  int e = idx >> 7;
  int f = idx & (HIDF - 1);
  int h = f >> 4;
  int s = src[e], d = dst[e];
  float w = ea[(size_t)e * NHEAD + h] / (denom[d * NHEAD + h] + 1e-16f);
  atomicAdd(&O[(size_t)d * HIDF + f], w * Hsrc[(size_t)s * HIDF + f]);
}

// ---------------------------------------------------------------------------
// Semantic score: score_r = dot(colmean_r, q); softmax over the 2 relations
// per destination type. colsum rows: 0=o_ba 1=o_aa 2=o_ab 3=o_bb.
// ---------------------------------------------------------------------------
__global__ void score_softmax(const float* __restrict__ colsum, const float* __restrict__ q,
                              float* __restrict__ attnw) {
  __shared__ float red[HIDF];
  __shared__ float sc[4];
  int f = threadIdx.x;   // 128 threads
  for (int r = 0; r < 4; ++r) {
    red[f] = colsum[r * HIDF + f] * (1.0f / (float)NNODES) * q[f];
    __syncthreads();
    for (int off = 64; off > 0; off >>= 1) {
      if (f < off) red[f] += red[f + off];
      __syncthreads();
    }
    if (f == 0) sc[r] = red[0];
    __syncthreads();
  }
  if (f == 0) {
    float m = fmaxf(sc[0], sc[1]);
    float e0 = __expf(sc[0] - m), e1 = __expf(sc[1] - m);
    attnw[0] = e0 / (e0 + e1); attnw[1] = e1 / (e0 + e1);   // type a
    m = fmaxf(sc[2], sc[3]);
    e0 = __expf(sc[2] - m); e1 = __expf(sc[3] - m);
    attnw[2] = e0 / (e0 + e1); attnw[3] = e1 / (e0 + e1);   // type b
  }
}

// ---------------------------------------------------------------------------
// Combine (ReLU folded) + BN batch statistics. Writes combined in place to O0.
// 256 threads = 2 rows x 128 features; grid-stride over rows.
// ---------------------------------------------------------------------------
__global__ void combine_stats(float* __restrict__ O0, const float* __restrict__ O1,
                              const float* __restrict__ attnw, int base,
                              float* __restrict__ bsum, float* __restrict__ bsq) {
  __shared__ float sh[256];
  int f = threadIdx.x & (HIDF - 1);
  int half = threadIdx.x >> 7;
  float w0 = attnw[base], w1 = attnw[base + 1];
  float s = 0.0f, ss = 0.0f;
  for (int row = blockIdx.x * 2 + half; row < NNODES; row += gridDim.x * 2) {
    size_t idx = (size_t)row * HIDF + f;
    float c = w0 * fmaxf(O0[idx], 0.0f) + w1 * fmaxf(O1[idx], 0.0f);
    O0[idx] = c;
    s += c; ss += c * c;
  }
  sh[threadIdx.x] = s;
  __syncthreads();
  float s2 = 0.0f;
  if (half == 0) s2 = sh[f] + sh[f + HIDF];
  __syncthreads();
  sh[threadIdx.x] = ss;
  __syncthreads();
  if (half == 0) {
    float ss2 = sh[f] + sh[f + HIDF];
    atomicAdd(&bsum[f], s2);
    atomicAdd(&bsq[f], ss2);
  }
}

// BatchNorm finalize (training mode: batch stats, biased variance).
__global__ void bn_final(const float* __restrict__ C, const float* __restrict__ bsum,
                         const float* __restrict__ bsq, const float* __restrict__ gamma,
                         const float* __restrict__ beta, float* __restrict__ out) {
  int idx = blockIdx.x * blockDim.x + threadIdx.x;
  if (idx >= NNODES * HIDF) return;
  int f = idx & (HIDF - 1);
  float mu = bsum[f] * (1.0f / (float)NNODES);
  float var = bsq[f] * (1.0f / (float)NNODES) - mu * mu;
  out[idx] = (C[idx] - mu) * rsqrtf(var + 1e-5f) * gamma[f] + beta[f];
}

// ---------------------------------------------------------------------------
extern "C" void kernel_launch(void* const* d_in, const int* in_sizes, int n_in,
                              void* d_out, int out_size, void* d_ws, size_t ws_size,
                              hipStream_t stream) {
  const float* x_a = (const float*)d_in[0];
  const float* x_b = (const float*)d_in[1];
  const int* ei_ab = (const int*)d_in[2];
  const int* ei_ba = (const int*)d_in[3];
  const int* ei_aa = (const int*)d_in[4];
  const int* ei_bb = (const int*)d_in[5];
  const float* W_a = (const float*)d_in[6];
  const float* b_a = (const float*)d_in[7];
  const float* W_b = (const float*)d_in[8];
  const float* b_b = (const float*)d_in[9];
  const float* att_ab_s = (const float*)d_in[10];
  const float* att_ab_d = (const float*)d_in[11];
  const float* att_ba_s = (const float*)d_in[12];
  const float* att_ba_d = (const float*)d_in[13];
  const float* att_aa_s = (const float*)d_in[14];
  const float* att_aa_d = (const float*)d_in[15];
  const float* att_bb_s = (const float*)d_in[16];
  const float* att_bb_d = (const float*)d_in[17];
  const float* Wk = (const float*)d_in[18];
  const float* bk = (const float*)d_in[19];
  const float* q  = (const float*)d_in[20];
  const float* gma = (const float*)d_in[21];
  const float* bta = (const float*)d_in[22];

  // ---- scratch layout ----
  char* w = (char*)d_ws;
  size_t off = 0;
  auto alloc = [&](size_t bytes) -> void* {
    void* p = w + off;
    off = (off + bytes + 255) & ~(size_t)255;
    return p;
  };
  const size_t NHbytes = (size_t)NNODES * HIDF * sizeof(float);   // 51.2 MB
  float* h_a  = (float*)alloc(NHbytes);
  float* h_b  = (float*)alloc(NHbytes);
  float* o_ba = (float*)alloc(NHbytes);   // dst type a
  float* o_aa = (float*)alloc(NHbytes);   // dst type a
  float* o_ab = (float*)alloc(NHbytes);   // dst type b
  float* o_bb = (float*)alloc(NHbytes);   // dst type b
  float* ea      = (float*)alloc((size_t)NEDGES * NHEAD * sizeof(float));    // reused
  unsigned* amax = (unsigned*)alloc((size_t)NNODES * NHEAD * sizeof(unsigned));
  float* denom   = (float*)alloc((size_t)NNODES * NHEAD * sizeof(float));
  float* as_ab = (float*)alloc((size_t)NNODES * NHEAD * sizeof(float));
  float* ad_ab = (float*)alloc((size_t)NNODES * NHEAD * sizeof(float));
  float* as_ba = (float*)alloc((size_t)NNODES * NHEAD * sizeof(float));
  float* ad_ba = (float*)alloc((size_t)NNODES * NHEAD * sizeof(float));
  float* as_aa = (float*)alloc((size_t)NNODES * NHEAD * sizeof(float));
  float* ad_aa = (float*)alloc((size_t)NNODES * NHEAD * sizeof(float));
  float* as_bb = (float*)alloc((size_t)NNODES * NHEAD * sizeof(float));
  float* ad_bb = (float*)alloc((size_t)NNODES * NHEAD * sizeof(float));
  float* colsum = (float*)alloc(4 * HIDF * sizeof(float));
  float* attnw  = (float*)alloc(4 * sizeof(float));
  float* bsum_a = (float*)alloc(HIDF * sizeof(float));
  float* bsq_a  = (float*)alloc(HIDF * sizeof(float));
  float* bsum_b = (float*)alloc(HIDF * sizeof(float));
  float* bsq_b  = (float*)alloc(HIDF * sizeof(float));

  // ---- zero accumulators (every call; graph-capture safe) ----
  (void)hipMemsetAsync(o_ba, 0, NHbytes, stream);
  (void)hipMemsetAsync(o_aa, 0, NHbytes, stream);
  (void)hipMemsetAsync(o_ab, 0, NHbytes, stream);
  (void)hipMemsetAsync(o_bb, 0, NHbytes, stream);
  (void)hipMemsetAsync(colsum, 0, 4 * HIDF * sizeof(float), stream);
  (void)hipMemsetAsync(bsum_a, 0, HIDF * sizeof(float), stream);
  (void)hipMemsetAsync(bsq_a, 0, HIDF * sizeof(float), stream);
  (void)hipMemsetAsync(bsum_b, 0, HIDF * sizeof(float), stream);
  (void)hipMemsetAsync(bsq_b, 0, HIDF * sizeof(float), stream);

  const int BLK = 256;
  const int ROWTILES = NNODES / 16;                        // 6250 (exact)
  const dim3 gemm_grid((ROWTILES + 7) / 8, HIDF / 16);     // 782 x 8
  const int na_grid = (NNODES * NHEAD + BLK - 1) / BLK;
  const int e_grid = (NEDGES + BLK - 1) / BLK;
  const int ef_grid = (NEDGES * HIDF + BLK - 1) / BLK;
  const int nf_grid = (NNODES * HIDF + BLK - 1) / BLK;

  // 1) projections (fp32 WMMA, B LDS-staged)
  proj_gemm<<<gemm_grid, BLK, 0, stream>>>(x_a, W_a, b_a, h_a);
  proj_gemm<<<gemm_grid, BLK, 0, stream>>>(x_b, W_b, b_b, h_b);

  // 2) node-level attention logit halves
  node_attn<<<na_grid, BLK, 0, stream>>>(h_a, att_ab_s, as_ab);
  node_attn<<<na_grid, BLK, 0, stream>>>(h_b, att_ab_d, ad_ab);
  node_attn<<<na_grid, BLK, 0, stream>>>(h_b, att_ba_s, as_ba);
  node_attn<<<na_grid, BLK, 0, stream>>>(h_a, att_ba_d, ad_ba);
  node_attn<<<na_grid, BLK, 0, stream>>>(h_a, att_aa_s, as_aa);
  node_attn<<<na_grid, BLK, 0, stream>>>(h_a, att_aa_d, ad_aa);
  node_attn<<<na_grid, BLK, 0, stream>>>(h_b, att_bb_s, as_bb);
  node_attn<<<na_grid, BLK, 0, stream>>>(h_b, att_bb_d, ad_bb);

  // 3) edge attention per relation (sequential -> L2-resident working set)
  struct Rel {
    const int* ei; const float* as_; const float* ad_; const float* hsrc; float* o;
  } rels[4] = {
    { ei_ab, as_ab, ad_ab, h_a, o_ab },   // a -> b
    { ei_bb, as_bb, ad_bb, h_b, o_bb },   // b -> b
    { ei_ba, as_ba, ad_ba, h_b, o_ba },   // b -> a
    { ei_aa, as_aa, ad_aa, h_a, o_aa },   // a -> a
  };
  for (int r = 0; r < 4; ++r) {
    const int* src = rels[r].ei;
    const int* dst = rels[r].ei + NEDGES;
    (void)hipMemsetAsync(amax, 0, (size_t)NNODES * NHEAD * sizeof(unsigned), stream);
    (void)hipMemsetAsync(denom, 0, (size_t)NNODES * NHEAD * sizeof(float), stream);
    edge_pass1<<<e_grid, BLK, 0, stream>>>(src, dst, rels[r].as_, rels[r].ad_, ea, amax);
    edge_pass2<<<e_grid, BLK, 0, stream>>>(dst, amax, ea, denom);
    edge_pass3<<<ef_grid, BLK, 0, stream>>>(src, dst, ea, denom, rels[r].hsrc, rels[r].o);
  }

  // 4) semantic attention column sums (fp32 WMMA; ReLU folded into A load)
  semantic_colsum<<<gemm_grid, BLK, 0, stream>>>(o_ba, Wk, bk, colsum + 0 * HIDF);
  semantic_colsum<<<gemm_grid, BLK, 0, stream>>>(o_aa, Wk, bk, colsum + 1 * HIDF);
  semantic_colsum<<<gemm_grid, BLK, 0, stream>>>(o_ab, Wk, bk, colsum + 2 * HIDF);
  semantic_colsum<<<gemm_grid, BLK, 0, stream>>>(o_bb, Wk, bk, colsum + 3 * HIDF);

  // 5) scores + softmax over relations per dst type
  score_softmax<<<1, HIDF, 0, stream>>>(colsum, q, attnw);

  // 6) combine (in place into o_ba / o_ab) + BN batch stats
  combine_stats<<<512, BLK, 0, stream>>>(o_ba, o_aa, attnw, 0, bsum_a, bsq_a);
  combine_stats<<<512, BLK, 0, stream>>>(o_ab, o_bb, attnw, 2, bsum_b, bsq_b);

  // 7) BN finalize -> d_out (out_a then out_b, flat)
  float* out_a = (float*)d_out;
  float* out_b = out_a + (size_t)NNODES * HIDF;
  bn_final<<<nf_grid, BLK, 0, stream>>>(o_ba, bsum_a, bsq_a, gma, bta, out_a);
  bn_final<<<nf_grid, BLK, 0, stream>>>(o_ab, bsum_b, bsq_b, gma, bta, out_b);
}